// ModernEdgeConv_59021440582229
// MI455X (gfx1250) — compile-verified
//
#include <hip/hip_runtime.h>
#include <math.h>

// ---------------- problem constants ----------------
#define BB   8
#define NN   4096
#define DD   64
#define KK   20
#define EE   129      // 2*D + 1
#define EP   132      // E padded to multiple of 4
#define HH   256      // 2*OUT
#define OO   128      // OUT

typedef __attribute__((ext_vector_type(2))) float v2f;
typedef __attribute__((ext_vector_type(8))) float v8f;
typedef __attribute__((ext_vector_type(4))) unsigned int v4u;
typedef __attribute__((ext_vector_type(8))) int v8i_;
typedef __attribute__((ext_vector_type(4))) int v4i_;

#if __has_builtin(__builtin_amdgcn_tensor_load_to_lds)
#define HAVE_TDM 1
#else
#define HAVE_TDM 0
#endif

#if HAVE_TDM
// Issue one TDM descriptor: copy a (tile1 x 16) f32 panel from a row-major
// (dim1 x dim0) tensor (panel column offset already folded into gaddr) into
// LDS at lds_off.  Rows >= dim1 are zero-filled by TDM OOB semantics.
__device__ __forceinline__ void tdm_load_panel(const void* gaddr, unsigned lds_off,
                                               unsigned dim0, unsigned dim1,
                                               unsigned tile1) {
    unsigned long long ga = (unsigned long long)(uintptr_t)gaddr;
    v4u g0 = { 1u,                                   // count=1, user descriptor
               lds_off,                              // LDS byte address
               (unsigned)(ga & 0xFFFFFFFFu),         // global_addr[31:0]
               (unsigned)((ga >> 32) & 0x1FFFFFFu) | (2u << 30) }; // [56:32]|type=2
    v8i_ g1 = { (int)0x00020000u,                                   // data_size=4B
                (int)((dim0 & 0xFFFFu) << 16),                      // tensor_dim0 lo
                (int)(((dim0 >> 16) & 0xFFFFu) | ((dim1 & 0xFFFFu) << 16)),
                (int)(((dim1 >> 16) & 0xFFFFu) | (16u << 16 << 0)), // tile_dim0=16
                (int)(tile1 & 0xFFFFu),                             // tile_dim1
                (int)dim0,                                          // dim0 stride
                0, 0 };
    v4i_ z = {0, 0, 0, 0};
#if __clang_major__ >= 23
    v8i_ z8 = {0, 0, 0, 0, 0, 0, 0, 0};
    __builtin_amdgcn_tensor_load_to_lds(g0, g1, z, z, z8, 0);
#else
    __builtin_amdgcn_tensor_load_to_lds(g0, g1, z, z, 0);
#endif
}
__device__ __forceinline__ void tdm_wait() {
#if __has_builtin(__builtin_amdgcn_s_wait_tensorcnt)
    __builtin_amdgcn_s_wait_tensorcnt(0);
#else
    asm volatile("s_wait_tensorcnt 0" ::: "memory");
#endif
}
#endif

// =====================================================================
// Kernel 1: per-point squared norms  sqn[b*N+n] = sum_d x^2
// =====================================================================
__global__ __launch_bounds__(256) void sqnorm_kernel(const float* __restrict__ x,
                                                     float* __restrict__ sqn) {
    int t = blockIdx.x * blockDim.x + threadIdx.x;   // 0 .. B*N-1
    const float4* xp = (const float4*)(x + (size_t)t * DD);
    float acc = 0.f;
#pragma unroll
    for (int i = 0; i < DD / 4; ++i) {
        float4 v = xp[i];
        acc += v.x * v.x + v.y * v.y + v.z * v.z + v.w * v.w;
    }
    sqn[t] = acc;
}

// =====================================================================
// Kernel 2: KNN (top-20 smallest distances) via f32 WMMA distance tiles
//   grid: B * (N/16) blocks, 256 threads (8 waves)
// =====================================================================
__global__ __launch_bounds__(256) void knn_kernel(const float* __restrict__ x,
                                                  const float* __restrict__ sqn,
                                                  int* __restrict__ knn) {
    __shared__ float Sq[16 * DD];        // 16 query rows
    __shared__ float Sqs[16];            // their sq norms
    __shared__ float Sd[16 * 128];       // distance chunk (16 rows x 128 cols)
    __shared__ float Smd[16][320];       // merge: 16 threads * 20 per row
    __shared__ int   Smi[16][320];

    const int blk  = blockIdx.x;
    const int b    = blk / (NN / 16);
    const int rb   = (blk % (NN / 16)) * 16;     // row base
    const int tid  = threadIdx.x;
    const int wave = tid >> 5;
    const int lane = tid & 31;
    const int lm   = lane & 15;
    const int koff = (lane < 16) ? 0 : 2;
    const int mhi  = (lane < 16) ? 0 : 8;

    // load 16 query rows (1024 floats) as float4s
    {
        const float4* src = (const float4*)(x + ((size_t)b * NN + rb) * DD);
        ((float4*)Sq)[tid] = src[tid];
        if (tid < 16) Sqs[tid] = sqn[(size_t)b * NN + rb + tid];
    }
    __syncthreads();

    // private top-20 (register resident via unrolled compare-select)
    float td[KK];
    int   ti[KK];
    float worst = 1e30f;
#pragma unroll
    for (int q = 0; q < KK; ++q) { td[q] = 1e30f; ti[q] = 0; }

    const int selrow = tid >> 4;   // 0..15
    const int selsub = tid & 15;   // 0..15 (8 cols each of 128)

    for (int cb = 0; cb < NN; cb += 128) {
        // ---- WMMA phase: wave w computes cols [cb+16w, cb+16w+16) ----
        const int col = cb + wave * 16 + lm;
        v8f c = 0;
        const float2* bcol = (const float2*)(x + ((size_t)b * NN + col) * DD);
#pragma unroll
        for (int s = 0; s < DD / 4; ++s) {
            const int k = 4 * s + koff;
            float2 af = *(const float2*)&Sq[lm * DD + k];
            float2 bf = bcol[k >> 1];
            v2f a = {af.x, af.y};
            v2f bv = {bf.x, bf.y};
            c = __builtin_amdgcn_wmma_f32_16x16x4_f32(false, a, false, bv,
                                                      (short)0, c, false, false);
        }
        const float scol = sqn[(size_t)b * NN + col];
#pragma unroll
        for (int r = 0; r < 8; ++r) {
            const int m = mhi + r;                    // local query row
            float d = Sqs[m] + scol - 2.0f * c[r];
            if (rb + m == col) d += 1e10f;            // mask self
            Sd[m * 128 + wave * 16 + lm] = d;
        }
        __syncthreads();

        // ---- selection phase: thread owns (row, 8 cols) of this chunk ----
#pragma unroll
        for (int j = 0; j < 8; ++j) {
            const int cc = selsub * 8 + j;
            const float d = Sd[selrow * 128 + cc];
            const int  gj = cb + cc;
            if (d < worst) {
                int wi = 0; float wv = td[0];
#pragma unroll
                for (int q = 1; q < KK; ++q)
                    if (td[q] > wv) { wv = td[q]; wi = q; }
#pragma unroll
                for (int q = 0; q < KK; ++q) {
                    const bool take = (q == wi);
                    td[q] = take ? d  : td[q];
                    ti[q] = take ? gj : ti[q];
                }
                float nw = td[0];
#pragma unroll
                for (int q = 1; q < KK; ++q) nw = fmaxf(nw, td[q]);
                worst = nw;
            }
        }
        __syncthreads();   // before next chunk overwrites Sd
    }

    // ---- merge 16 per-thread lists per row ----
#pragma unroll
    for (int q = 0; q < KK; ++q) {
        Smd[selrow][selsub * KK + q] = td[q];
        Smi[selrow][selsub * KK + q] = ti[q];
    }
    __syncthreads();

    if (selsub == 0) {
        int* dst = knn + ((size_t)b * NN + rb + selrow) * KK;
        for (int kk = 0; kk < KK; ++kk) {
            float best = 1e30f; int bi = 0; int bslot = 0;
            for (int s = 0; s < 320; ++s) {
                const float v = Smd[selrow][s];
                if (v < best) { best = v; bi = Smi[selrow][s]; bslot = s; }
            }
            Smd[selrow][bslot] = 1e30f;
            dst[kk] = bi;
        }
    }
}

// =====================================================================
// Kernel 3: edge features + LayerNorm + MLP (WMMA) + max over K
//   grid: B * (N/4) blocks, 160 threads (5 waves)
//   block handles 4 points -> 80 edge rows = 5 M-tiles of 16
//   Weight panels (16 cols) staged into LDS via the Tensor Data Mover.
// =====================================================================
__global__ __launch_bounds__(160) void edge_mlp_kernel(
        const float* __restrict__ x, const int* __restrict__ knn,
        const float* __restrict__ lns, const float* __restrict__ lnb,
        const float* __restrict__ w1, const float* __restrict__ b1,
        const float* __restrict__ w2, const float* __restrict__ b2,
        float* __restrict__ out) {
    __shared__ float Sedge[80 * EP];     // edge features; later aliased as h2[80*128]
    __shared__ float Sh1[80 * HH];       // swish(layer1)
    __shared__ float Swp[HH * 16];       // staged weight panel (max 256x16)
    __shared__ float Shalf[160];         // dist_sq partials
    float* Sh2 = Sedge;                  // alias (edge dead after layer1)

    const int blk   = blockIdx.x;
    const int b     = blk / (NN / 4);
    const int pbase = (blk % (NN / 4)) * 4;
    const int tid   = threadIdx.x;

    // ---------- phase 1: build 80 edge rows + LayerNorm ----------
    {
        const int e    = tid >> 1;        // 0..79
        const int half = tid & 1;
        const int p    = e / KK;
        const int kk   = e % KK;
        const int pt   = pbase + p;
        const int nb   = knn[((size_t)b * NN + pt) * KK + kk];
        const float* xc = x + ((size_t)b * NN + pt) * DD;
        const float* xn = x + ((size_t)b * NN + nb) * DD;
        float acc = 0.f;
#pragma unroll
        for (int i = 0; i < 32; ++i) {
            const int d = half * 32 + i;
            const float c0 = xc[d];
            const float df = xn[d] - c0;
            Sedge[e * EP + d]      = c0;
            Sedge[e * EP + DD + d] = df;
            acc += df * df;
        }
        Shalf[e * 2 + half] = acc;
        __syncthreads();
        if (half == 0) {
            const float dist = Shalf[e * 2] + Shalf[e * 2 + 1];
            float* row = Sedge + e * EP;
            row[128] = dist;
            row[129] = 0.f; row[130] = 0.f; row[131] = 0.f;
            float mu = 0.f;
            for (int c = 0; c < EE; ++c) mu += row[c];
            mu *= (1.0f / EE);
            float var = 0.f;
            for (int c = 0; c < EE; ++c) { const float t = row[c] - mu; var += t * t; }
            var *= (1.0f / EE);
            const float rs = rsqrtf(var + 1e-6f);
            for (int c = 0; c < EE; ++c)
                row[c] = (row[c] - mu) * rs * lns[c] + lnb[c];
        }
        __syncthreads();
    }

    const int wave = tid >> 5;            // 0..4  -> M-tile
    const int lane = tid & 31;
    const int lm   = lane & 15;
    const int koff = (lane < 16) ? 0 : 2;
    const int mhi  = (lane < 16) ? 0 : 8;
    const int arow = wave * 16 + lm;      // A-operand row for this lane

    // ---------- phase 2: layer1  h1 = swish(edge @ w1 + b1) ----------
    for (int nt = 0; nt < HH / 16; ++nt) {
        // stage w1 panel [0..131][nt*16 .. nt*16+16) into Swp (rows>=129 -> 0)
#if HAVE_TDM
        if (tid == 0)
            tdm_load_panel(w1 + nt * 16, (unsigned)(uintptr_t)(void*)Swp,
                           HH, EE, EP);
        tdm_wait();
#else
        for (int i = tid; i < EP * 16; i += 160) {
            const int row = i >> 4, cc = i & 15;
            Swp[i] = (row < EE) ? w1[(size_t)row * HH + nt * 16 + cc] : 0.f;
        }
#endif
        __syncthreads();

        const int n = nt * 16 + lm;
        v8f c = 0;
        for (int s = 0; s < EP / 4; ++s) {            // 33 K-steps
            const int k = 4 * s + koff;
            float2 af = *(const float2*)&Sedge[arow * EP + k];
            v2f a = {af.x, af.y};
            v2f bv = {Swp[k * 16 + lm], Swp[(k + 1) * 16 + lm]};
            c = __builtin_amdgcn_wmma_f32_16x16x4_f32(false, a, false, bv,
                                                      (short)0, c, false, false);
        }
        const float bias = b1[n];
#pragma unroll
        for (int r = 0; r < 8; ++r) {
            const int m = wave * 16 + mhi + r;
            float v = c[r] + bias;
            v = v / (1.0f + __expf(-v));              // swish
            Sh1[m * HH + n] = v;
        }
        __syncthreads();                              // panel reads done
    }

    // ---------- phase 3: layer2  h2 = h1 @ w2 + b2 ----------
    for (int nt = 0; nt < OO / 16; ++nt) {
#if HAVE_TDM
        if (tid == 0)
            tdm_load_panel(w2 + nt * 16, (unsigned)(uintptr_t)(void*)Swp,
                           OO, HH, HH);
        tdm_wait();
#else
        for (int i = tid; i < HH * 16; i += 160) {
            const int row = i >> 4, cc = i & 15;
            Swp[i] = w2[(size_t)row * OO + nt * 16 + cc];
        }
#endif
        __syncthreads();

        const int n = nt * 16 + lm;
        v8f c = 0;
        for (int s = 0; s < HH / 4; ++s) {            // 64 K-steps
            const int k = 4 * s + koff;
            float2 af = *(const float2*)&Sh1[arow * HH + k];
            v2f a = {af.x, af.y};
            v2f bv = {Swp[k * 16 + lm], Swp[(k + 1) * 16 + lm]};
            c = __builtin_amdgcn_wmma_f32_16x16x4_f32(false, a, false, bv,
                                                      (short)0, c, false, false);
        }
        const float bias = b2[n];
#pragma unroll
        for (int r = 0; r < 8; ++r) {
            const int m = wave * 16 + mhi + r;
            Sh2[m * OO + n] = c[r] + bias;
        }
        __syncthreads();
    }
    __syncthreads();

    // ---------- phase 4: max over K=20, write out ----------
    for (int o = tid; o < 4 * OO; o += 160) {
        const int p   = o >> 7;
        const int col = o & (OO - 1);
        const int m0  = p * KK;
        float mv = Sh2[m0 * OO + col];
#pragma unroll
        for (int q = 1; q < KK; ++q)
            mv = fmaxf(mv, Sh2[(m0 + q) * OO + col]);
        out[((size_t)b * NN + pbase + p) * OO + col] = mv;
    }
}

// =====================================================================
// host-side launcher
// =====================================================================
extern "C" void kernel_launch(void* const* d_in, const int* in_sizes, int n_in,
                              void* d_out, int out_size, void* d_ws, size_t ws_size,
                              hipStream_t stream) {
    const float* x   = (const float*)d_in[0];
    const float* lns = (const float*)d_in[1];
    const float* lnb = (const float*)d_in[2];
    const float* w1  = (const float*)d_in[3];
    const float* b1  = (const float*)d_in[4];
    const float* w2  = (const float*)d_in[5];
    const float* b2  = (const float*)d_in[6];
    float* out = (float*)d_out;

    float* sqn = (float*)d_ws;
    int*   knn = (int*)((char*)d_ws + (size_t)BB * NN * sizeof(float));

    sqnorm_kernel<<<(BB * NN) / 256, 256, 0, stream>>>(x, sqn);
    knn_kernel<<<BB * (NN / 16), 256, 0, stream>>>(x, sqn, knn);
    edge_mlp_kernel<<<BB * (NN / 4), 160, 0, stream>>>(x, knn, lns, lnb,
                                                       w1, b1, w2, b2, out);
}